// UVMDBlock_44461501448531
// MI455X (gfx1250) — compile-verified
//
#include <hip/hip_runtime.h>
#include <math.h>

// ---------------------------------------------------------------------------
// UVMD block for MI455X (gfx1250): four-step FFT (4096 = 64x64) where the
// 64-point DFT stages are fp32 GEMMs on V_WMMA_F32_16X16X4_F32.
// ---------------------------------------------------------------------------

typedef float v2f __attribute__((ext_vector_type(2)));
typedef float v8f __attribute__((ext_vector_type(8)));

#define S1 2064          // padded bin stride (2049 -> 2064, multiple of 16)
#define LDA 68           // padded LDS matrix stride (64 + 4): conflict-free cols
#define TWO_PI_64 0.09817477042468103f
#define TWO_PI_4096 0.0015339807878856412f

static __device__ __forceinline__ v8f wmma4(v2f a, v2f b, v8f c) {
  // D = A(16x4) * B(4x16) + C, fp32. neg_a/neg_b must be 0 for f32 per ISA.
  return __builtin_amdgcn_wmma_f32_16x16x4_f32(false, a, false, b,
                                               (short)0, c, false, false);
}

// A-fragment (16x4) from LDS plane p at tile (m0,k0).
static __device__ __forceinline__ v2f ldsA(const float* p, int m0, int k0,
                                           int lm, int half) {
  const float* r = p + (m0 + lm) * LDA + k0 + 2 * half;
  v2f a; a.x = r[0]; a.y = r[1]; return a;
}

// B-fragment (4x16) from LDS plane p at tile (k0,n0).
static __device__ __forceinline__ v2f ldsB(const float* p, int k0, int n0,
                                           int lm, int half) {
  int r0 = k0 + 2 * half;
  v2f b;
  b.x = p[r0 * LDA + n0 + lm];
  b.y = p[(r0 + 1) * LDA + n0 + lm];
  return b;
}

// Analytic DFT-64 fragment: element j has phase other*(kbase+j) mod 64.
// Works for A-fragments (other = m0+lm) and B-fragments (other = n0+lm).
static __device__ __forceinline__ void dftFrag(int other, int kbase,
                                               v2f& c, v2f& s) {
  int p0 = (other * kbase) & 63;
  int p1 = (other * (kbase + 1)) & 63;
  float s0, c0, s1, c1;
  __sincosf((float)p0 * TWO_PI_64, &s0, &c0);
  __sincosf((float)p1 * TWO_PI_64, &s1, &c1);
  c.x = c0; c.y = c1; s.x = s0; s.y = s1;
}

// ---------------------------------------------------------------------------
// Kernel 1: forward rfft per row. Row r = b*16+c reads x[b, :, c].
// X[n1][n2] = x[64*n1+n2]; Y = W64*X; twiddle; R = Z*W64; bin = k1 + 64*k2.
// ---------------------------------------------------------------------------
__global__ __launch_bounds__(256) void fwd_fft_kernel(
    const float* __restrict__ x, float* __restrict__ FR,
    float* __restrict__ FI) {
  __shared__ float smem[2 * 64 * LDA];
  float* P0 = smem;
  float* P1 = smem + 64 * LDA;

  const int tid = threadIdx.x;
  const int lane = tid & 31, wave = tid >> 5;
  const int lm = lane & 15, half = lane >> 4;
  const int row = blockIdx.x;
  const int b = row >> 4, c = row & 15;
  const float* xrow = x + (size_t)b * 4096 * 16 + c;

  // Load 4096 reals into P0 as 64x64 (row n1, col n2).
#pragma unroll
  for (int j = 0; j < 16; ++j) {
    int t = tid + 256 * j;
    P0[(t >> 6) * LDA + (t & 63)] = xrow[(size_t)t * 16];
  }
  __syncthreads();

  // Stage 1: Yr = cos*X, Yi = -sin*X. Keep both tiles in registers.
  v8f yr[2], yi[2];
#pragma unroll
  for (int tI = 0; tI < 2; ++tI) {
    int tile = wave + 8 * tI;
    int m0 = (tile >> 2) * 16, n0 = (tile & 3) * 16;
    v8f ar = {0.f, 0.f, 0.f, 0.f, 0.f, 0.f, 0.f, 0.f};
    v8f ai = ar;
    for (int k0 = 0; k0 < 64; k0 += 4) {
      v2f xb = ldsB(P0, k0, n0, lm, half);
      v2f cf, sf;
      dftFrag(m0 + lm, k0 + 2 * half, cf, sf);
      v2f nsf; nsf.x = -sf.x; nsf.y = -sf.y;
      ar = wmma4(cf, xb, ar);
      ai = wmma4(nsf, xb, ai);
    }
    yr[tI] = ar; yi[tI] = ai;
  }
  __syncthreads();  // all reads of X done -> reuse LDS for Y

#pragma unroll
  for (int tI = 0; tI < 2; ++tI) {
    int tile = wave + 8 * tI;
    int m0 = (tile >> 2) * 16, n0 = (tile & 3) * 16;
#pragma unroll
    for (int r = 0; r < 8; ++r) {
      int mm = m0 + r + 8 * half, nn = n0 + lm;
      P0[mm * LDA + nn] = yr[tI][r];
      P1[mm * LDA + nn] = yi[tI][r];
    }
  }
  __syncthreads();

  // Twiddle: Z[k1][n2] = Y * exp(-2*pi*i*k1*n2/4096).
#pragma unroll
  for (int j = 0; j < 16; ++j) {
    int e = tid + 256 * j;
    int k1 = e >> 6, n2 = e & 63;
    float sn, cs;
    __sincosf((float)((k1 * n2) & 4095) * TWO_PI_4096, &sn, &cs);
    float a = P0[k1 * LDA + n2], bb = P1[k1 * LDA + n2];
    P0[k1 * LDA + n2] = a * cs + bb * sn;
    P1[k1 * LDA + n2] = bb * cs - a * sn;
  }
  __syncthreads();

  // Stage 2: R = Z * W64. Rr = Zr*cos + Zi*sin ; Ri = -Zr*sin + Zi*cos.
#pragma unroll
  for (int tI = 0; tI < 2; ++tI) {
    int tile = wave + 8 * tI;
    int m0 = (tile >> 2) * 16, n0 = (tile & 3) * 16;
    v8f ar = {0.f, 0.f, 0.f, 0.f, 0.f, 0.f, 0.f, 0.f};
    v8f ai = ar;
    for (int k0 = 0; k0 < 64; k0 += 4) {
      v2f zr = ldsA(P0, m0, k0, lm, half);
      v2f zi = ldsA(P1, m0, k0, lm, half);
      v2f cf, sf;
      dftFrag(n0 + lm, k0 + 2 * half, cf, sf);
      v2f nsf; nsf.x = -sf.x; nsf.y = -sf.y;
      ar = wmma4(zr, cf, ar);
      ar = wmma4(zi, sf, ar);
      ai = wmma4(zr, nsf, ai);
      ai = wmma4(zi, cf, ai);
    }
#pragma unroll
    for (int r = 0; r < 8; ++r) {
      int k1 = m0 + r + 8 * half, k2 = n0 + lm;
      int bin = k1 + 64 * k2;
      if (bin <= 2048) {
        FR[(size_t)row * S1 + bin] = ar[r];
        FI[(size_t)row * S1 + bin] = ai[r];
      }
    }
  }
}

// ---------------------------------------------------------------------------
// Kernel 2: 8 Jacobi iterations over K=8 modes, per (row, bin), in registers.
// ---------------------------------------------------------------------------
__global__ __launch_bounds__(256) void iterate_kernel(
    const float* __restrict__ FR, const float* __restrict__ FI,
    const float* __restrict__ log_alpha, const float* __restrict__ raw_tau,
    const float* __restrict__ raw_omega, float* __restrict__ UR,
    float* __restrict__ UI) {
  int bin = blockIdx.x * 256 + threadIdx.x;
  int row = blockIdx.y;
  if (bin > 2048) return;

  float fr = FR[(size_t)row * S1 + bin];
  float fi = FI[(size_t)row * S1 + bin];
  float freq = 0.5f * (float)bin * (1.0f / 2048.0f);

  float omega[8];
#pragma unroll
  for (int k = 0; k < 8; ++k)
    omega[k] = 0.5f / (1.0f + __expf(-raw_omega[k]));

  float ur[8], ui[8];
#pragma unroll
  for (int k = 0; k < 8; ++k) { ur[k] = 0.f; ui[k] = 0.f; }
  float lr = 0.f, li = 0.f;

  for (int l = 0; l < 8; ++l) {
    float tl = __logf(1.0f + __expf(raw_tau[l]));  // softplus
    float sr = 0.f, si = 0.f;
#pragma unroll
    for (int k = 0; k < 8; ++k) { sr += ur[k]; si += ui[k]; }
    float nr_sum = 0.f, ni_sum = 0.f;
    float un_r[8], un_i[8];
#pragma unroll
    for (int k = 0; k < 8; ++k) {
      float al = __expf(log_alpha[l * 8 + k]);
      float d = freq - omega[k];
      float den = 1.0f + 2.0f * al * d * d;
      float inv = 1.0f / den;
      float nr = fr - (sr - ur[k]) + 0.5f * lr;
      float ni = fi - (si - ui[k]) + 0.5f * li;
      un_r[k] = nr * inv; un_i[k] = ni * inv;
      nr_sum += un_r[k]; ni_sum += un_i[k];
    }
    lr += tl * (fr - nr_sum);
    li += tl * (fi - ni_sum);
#pragma unroll
    for (int k = 0; k < 8; ++k) { ur[k] = un_r[k]; ui[k] = un_i[k]; }
  }

#pragma unroll
  for (int k = 0; k < 8; ++k) {
    size_t o = ((size_t)(k * 1024 + row)) * S1 + bin;
    UR[o] = ur[k];
    UI[o] = ui[k];
  }
}

// ---------------------------------------------------------------------------
// Kernel 3: irfft per (row, mode). Hermitian-extend into 64x64, conjugate
// four-step FFT (real part of output only), scale 1/4096, write transposed.
// ---------------------------------------------------------------------------
__global__ __launch_bounds__(256) void inv_fft_kernel(
    const float* __restrict__ UR, const float* __restrict__ UI,
    float* __restrict__ out) {
  __shared__ float smem[2 * 64 * LDA];
  float* P0 = smem;
  float* P1 = smem + 64 * LDA;

  const int tid = threadIdx.x;
  const int lane = tid & 31, wave = tid >> 5;
  const int lm = lane & 15, half = lane >> 4;
  const int rowk = blockIdx.x;
  const int kmode = rowk >> 10;
  const int row = rowk & 1023;
  const int b = row >> 4, c = row & 15;

  // Hermitian reconstruction of full 4096-bin spectrum into LDS.
#pragma unroll
  for (int j = 0; j < 16; ++j) {
    int s = tid + 256 * j;
    float vr, vi;
    if (s <= 2048) {
      vr = UR[(size_t)rowk * S1 + s];
      vi = UI[(size_t)rowk * S1 + s];
    } else {
      int m = 4096 - s;
      vr = UR[(size_t)rowk * S1 + m];
      vi = -UI[(size_t)rowk * S1 + m];
    }
    P0[(s >> 6) * LDA + (s & 63)] = vr;
    P1[(s >> 6) * LDA + (s & 63)] = vi;
  }
  __syncthreads();

  // Stage 1 (conjugate DFT): Yr = cos*Ur - sin*Ui ; Yi = cos*Ui + sin*Ur.
  v8f yr[2], yi[2];
#pragma unroll
  for (int tI = 0; tI < 2; ++tI) {
    int tile = wave + 8 * tI;
    int m0 = (tile >> 2) * 16, n0 = (tile & 3) * 16;
    v8f ar = {0.f, 0.f, 0.f, 0.f, 0.f, 0.f, 0.f, 0.f};
    v8f ai = ar;
    for (int k0 = 0; k0 < 64; k0 += 4) {
      v2f br = ldsB(P0, k0, n0, lm, half);
      v2f bi = ldsB(P1, k0, n0, lm, half);
      v2f cf, sf;
      dftFrag(m0 + lm, k0 + 2 * half, cf, sf);
      v2f nsf; nsf.x = -sf.x; nsf.y = -sf.y;
      ar = wmma4(cf, br, ar);
      ar = wmma4(nsf, bi, ar);
      ai = wmma4(cf, bi, ai);
      ai = wmma4(sf, br, ai);
    }
    yr[tI] = ar; yi[tI] = ai;
  }
  __syncthreads();

#pragma unroll
  for (int tI = 0; tI < 2; ++tI) {
    int tile = wave + 8 * tI;
    int m0 = (tile >> 2) * 16, n0 = (tile & 3) * 16;
#pragma unroll
    for (int r = 0; r < 8; ++r) {
      int mm = m0 + r + 8 * half, nn = n0 + lm;
      P0[mm * LDA + nn] = yr[tI][r];
      P1[mm * LDA + nn] = yi[tI][r];
    }
  }
  __syncthreads();

  // Twiddle (+i sin): Zr = Yr*cs - Yi*sn ; Zi = Yi*cs + Yr*sn.
#pragma unroll
  for (int j = 0; j < 16; ++j) {
    int e = tid + 256 * j;
    int k1 = e >> 6, n2 = e & 63;
    float sn, cs;
    __sincosf((float)((k1 * n2) & 4095) * TWO_PI_4096, &sn, &cs);
    float a = P0[k1 * LDA + n2], bb = P1[k1 * LDA + n2];
    P0[k1 * LDA + n2] = a * cs - bb * sn;
    P1[k1 * LDA + n2] = bb * cs + a * sn;
  }
  __syncthreads();

  // Stage 2 (real part only): Rr = Zr*cos - Zi*sin. t = k1 + 64*k2.
#pragma unroll
  for (int tI = 0; tI < 2; ++tI) {
    int tile = wave + 8 * tI;
    int m0 = (tile >> 2) * 16, n0 = (tile & 3) * 16;
    v8f ar = {0.f, 0.f, 0.f, 0.f, 0.f, 0.f, 0.f, 0.f};
    for (int k0 = 0; k0 < 64; k0 += 4) {
      v2f zr = ldsA(P0, m0, k0, lm, half);
      v2f zi = ldsA(P1, m0, k0, lm, half);
      v2f cf, sf;
      dftFrag(n0 + lm, k0 + 2 * half, cf, sf);
      v2f nsf; nsf.x = -sf.x; nsf.y = -sf.y;
      ar = wmma4(zr, cf, ar);
      ar = wmma4(zi, nsf, ar);
    }
#pragma unroll
    for (int r = 0; r < 8; ++r) {
      int k1 = m0 + r + 8 * half, k2 = n0 + lm;
      int t = k1 + 64 * k2;
      size_t o = ((((size_t)b * 8 + kmode) * 4096) + t) * 16 + c;
      out[o] = ar[r] * (1.0f / 4096.0f);
    }
  }
}

// ---------------------------------------------------------------------------
extern "C" void kernel_launch(void* const* d_in, const int* in_sizes, int n_in,
                              void* d_out, int out_size, void* d_ws,
                              size_t ws_size, hipStream_t stream) {
  (void)in_sizes; (void)n_in; (void)out_size; (void)ws_size;
  const float* x = (const float*)d_in[0];          // (64, 4096, 16)
  const float* log_alpha = (const float*)d_in[1];  // (8, 8)
  const float* raw_tau = (const float*)d_in[2];    // (8,)
  const float* raw_omega = (const float*)d_in[3];  // (8,)

  float* ws = (float*)d_ws;
  float* FR = ws;                       // 1024 * S1
  float* FI = FR + (size_t)1024 * S1;   // 1024 * S1
  float* UR = FI + (size_t)1024 * S1;   // 8192 * S1
  float* UI = UR + (size_t)8192 * S1;   // 8192 * S1

  fwd_fft_kernel<<<1024, 256, 0, stream>>>(x, FR, FI);
  dim3 g2(9, 1024);
  iterate_kernel<<<g2, 256, 0, stream>>>(FR, FI, log_alpha, raw_tau, raw_omega,
                                         UR, UI);
  inv_fft_kernel<<<8192, 256, 0, stream>>>(UR, UI, (float*)d_out);
}